// SelfAttention_65085934403863
// MI455X (gfx1250) — compile-verified
//
#include <hip/hip_runtime.h>
#include <hip/hip_bf16.h>
#include <math.h>

// Problem constants (match reference)
#define BH 16      // heads
#define TT 2048    // sequence length
#define DD 64      // head dim
#define WPB 4      // waves per block

static constexpr float TAU_RC  = 0.02f;
static constexpr float TAU_REF = 0.002f;
static constexpr float V_TH    = 0.01f;
static constexpr float EPS_T   = 1e-7f;
static constexpr float SIGMA   = 1e-3f;

typedef __attribute__((ext_vector_type(2))) float v2f;
typedef __attribute__((ext_vector_type(8))) float v8f;

// ---------------------------------------------------------------------------
// LIF rate nonlinearity (already divided by temperature sqrt(64)=8)
// ---------------------------------------------------------------------------
__device__ __forceinline__ float lif_rate(float dot, float g, float b) {
    float I = g * dot + b;
    float r = 0.f;
    if (I > (V_TH + EPS_T)) {
        float lt  = log1pf(-V_TH / I);        // < 0
        float den = TAU_REF - TAU_RC * lt;    // > TAU_REF
        r = 1.0f / den;
        if (!(r >= 0.f)) r = 0.f;             // nan_to_num guard
    }
    return r * 0.125f;                        // 1/sqrt(D)
}

// ---------------------------------------------------------------------------
// Kernel 1: reciprocal L2 norms of K rows -> d_ws  (rnorm[h*T + t])
// ---------------------------------------------------------------------------
__global__ __launch_bounds__(256) void knorm_kernel(const float* __restrict__ K,
                                                    float* __restrict__ rnorm) {
    int i = blockIdx.x * blockDim.x + threadIdx.x;
    if (i >= BH * TT) return;
    const float4* kr = (const float4*)(K + (size_t)i * DD);
    float s = 0.f;
#pragma unroll
    for (int j = 0; j < DD / 4; ++j) {
        float4 k4 = kr[j];
        s += k4.x * k4.x + k4.y * k4.y + k4.z * k4.z + k4.w * k4.w;
    }
    float n = sqrtf(s);
    rnorm[i] = 1.0f / fmaxf(n, 1e-8f);
}

// ---------------------------------------------------------------------------
// Kernel 2: one wave32 per 32-query block (two 16x16 M-tiles). V fragments
// for the key tile are batch-loaded at the top of the iteration so their
// latency hides under the QK WMMA chain + LIF elementwise work.
// ---------------------------------------------------------------------------
__global__ __launch_bounds__(WPB * 32) void lif_attn_kernel(
    const float* __restrict__ Q, const float* __restrict__ K,
    const float* __restrict__ V, const float* __restrict__ gain,
    const float* __restrict__ bias, const float* __restrict__ rnorm,
    float* __restrict__ out) {
    __shared__ float lds[WPB][2][16][17];  // per-wave rates slabs (padded)

    const int lane = threadIdx.x & 31;
    const int wave = threadIdx.x >> 5;
    const int nTiles = BH * (TT / 32);                 // 1024
    const int wg   = blockIdx.x * WPB + wave;
    const int idx  = nTiles - 1 - wg;                  // longest work first
    const int h    = idx >> 6;                         // head (64 blocks/head)
    const int qb0  = (idx & 63) << 5;                  // 32-query block start
    const int half = lane >> 4;                        // 0 | 1
    const int l16  = lane & 15;

    const float* Qh = Q + (size_t)h * TT * DD;
    const float* Kh = K + (size_t)h * TT * DD;
    const float* Vh = V + (size_t)h * TT * DD;
    const float* rn = rnorm + h * TT;
    float*       Oh = out + (size_t)h * TT * DD;

    // --- Preload both Q A-tiles: qa[t][kk] = Q[m = qb0+t*16+l16, kk*4+half*2+{0,1}]
    v2f qa[2][16];
#pragma unroll
    for (int t = 0; t < 2; ++t) {
        const float* qrow = Qh + (size_t)(qb0 + t * 16 + l16) * DD + half * 2;
#pragma unroll
        for (int kk = 0; kk < 16; ++kk)
            qa[t][kk] = *(const v2f*)(qrow + kk * 4);
    }

    // --- Per-row gain/bias for rows held by this lane half (m = v + 8*half)
    float garr[2][8], barr[2][8];
#pragma unroll
    for (int t = 0; t < 2; ++t)
#pragma unroll
        for (int v = 0; v < 8; ++v) {
            int qrow = qb0 + t * 16 + v + half * 8;
            garr[t][v] = gain[h * TT + qrow];
            barr[t][v] = bias[h * TT + qrow];
        }

    const v8f zero8 = {0.f, 0.f, 0.f, 0.f, 0.f, 0.f, 0.f, 0.f};
    v8f acc[2][4];                 // per M-tile: 4 N-tiles covering D=64
#pragma unroll
    for (int t = 0; t < 2; ++t)
#pragma unroll
        for (int nt = 0; nt < 4; ++nt) acc[t][nt] = zero8;
    float ssq[2][8];
#pragma unroll
    for (int t = 0; t < 2; ++t)
#pragma unroll
        for (int v = 0; v < 8; ++v) ssq[t][v] = 0.f;

    // =============== main loop: key tiles valid for BOTH M-tiles =============
    for (int s0 = 0; s0 <= qb0; s0 += 16) {
        const float  krn  = rn[s0 + l16];
        const float* krow = Kh + (size_t)(s0 + l16) * DD + half * 2;
        __builtin_prefetch(krow + 16 * DD, 0, 3);      // next K tile
        __builtin_prefetch(Vh + (size_t)(s0 + 16 + half * 2) * DD + l16, 0, 3);

        // ---- batch-load ALL V B-fragments for this key tile up front -------
        // vfrag[nt*4+kk2]: rows s0 + kk2*4 + half*2 + {0,1}, col nt*16 + l16
        v2f vfrag[16];
#pragma unroll
        for (int nt = 0; nt < 4; ++nt) {
#pragma unroll
            for (int kk2 = 0; kk2 < 4; ++kk2) {
                const float* vp = Vh + (size_t)(s0 + kk2 * 4 + half * 2) * DD
                                     + nt * 16 + l16;
                vfrag[nt * 4 + kk2].x = vp[0];
                vfrag[nt * 4 + kk2].y = vp[DD];
            }
        }

        // ---- QK_hat^T: 16 K-steps, each B-fragment feeds 2 WMMAs ----
        v8f ct[2];
        ct[0] = zero8;
        ct[1] = zero8;
#pragma unroll
        for (int kk = 0; kk < 16; ++kk) {
            v2f b = *(const v2f*)(krow + kk * 4);
            b *= krn;                                  // fold K normalization
            ct[0] = __builtin_amdgcn_wmma_f32_16x16x4_f32(
                        false, qa[0][kk], false, b, (short)0, ct[0], false, false);
            ct[1] = __builtin_amdgcn_wmma_f32_16x16x4_f32(
                        false, qa[1][kk], false, b, (short)0, ct[1], false, false);
        }

        // ---- LIF + causal mask + ssq, stage rates to LDS ----
#pragma unroll
        for (int t = 0; t < 2; ++t) {
            const bool diag = (t == 0) && (s0 == qb0);
#pragma unroll
            for (int v = 0; v < 8; ++v) {
                int   m = v + half * 8;
                float r = lif_rate(ct[t][v], garr[t][v], barr[t][v]);
                if (diag && (l16 > m)) r = 0.f;
                ssq[t][v] += r * r;
                lds[wave][t][m][l16] = r;
            }
        }
        asm volatile("s_wait_dscnt 0x0" ::: "memory"); // same-wave RAW fence

        v2f a2[2][4];
#pragma unroll
        for (int t = 0; t < 2; ++t) {
            const float* lr = &lds[wave][t][l16][0];
#pragma unroll
            for (int kk2 = 0; kk2 < 4; ++kk2) {
                a2[t][kk2].x = lr[kk2 * 4 + half * 2];
                a2[t][kk2].y = lr[kk2 * 4 + half * 2 + 1];
            }
        }

        // ---- rates @ V: back-to-back WMMAs on resident fragments ----
#pragma unroll
        for (int nt = 0; nt < 4; ++nt) {
#pragma unroll
            for (int kk2 = 0; kk2 < 4; ++kk2) {
                v2f b = vfrag[nt * 4 + kk2];
                acc[0][nt] = __builtin_amdgcn_wmma_f32_16x16x4_f32(
                                 false, a2[0][kk2], false, b, (short)0,
                                 acc[0][nt], false, false);
                acc[1][nt] = __builtin_amdgcn_wmma_f32_16x16x4_f32(
                                 false, a2[1][kk2], false, b, (short)0,
                                 acc[1][nt], false, false);
            }
        }
    }

    // =============== epilogue: diagonal key tile for M-tile 1 only ===========
    {
        const int    s0   = qb0 + 16;
        const float  krn  = rn[s0 + l16];
        const float* krow = Kh + (size_t)(s0 + l16) * DD + half * 2;

        v2f vfrag[16];
#pragma unroll
        for (int nt = 0; nt < 4; ++nt) {
#pragma unroll
            for (int kk2 = 0; kk2 < 4; ++kk2) {
                const float* vp = Vh + (size_t)(s0 + kk2 * 4 + half * 2) * DD
                                     + nt * 16 + l16;
                vfrag[nt * 4 + kk2].x = vp[0];
                vfrag[nt * 4 + kk2].y = vp[DD];
            }
        }

        v8f c1 = zero8;
#pragma unroll
        for (int kk = 0; kk < 16; ++kk) {
            v2f b = *(const v2f*)(krow + kk * 4);
            b *= krn;
            c1 = __builtin_amdgcn_wmma_f32_16x16x4_f32(
                     false, qa[1][kk], false, b, (short)0, c1, false, false);
        }
#pragma unroll
        for (int v = 0; v < 8; ++v) {
            int   m = v + half * 8;
            float r = lif_rate(c1[v], garr[1][v], barr[1][v]);
            if (l16 > m) r = 0.f;                      // diagonal mask
            ssq[1][v] += r * r;
            lds[wave][1][m][l16] = r;
        }
        asm volatile("s_wait_dscnt 0x0" ::: "memory");

        v2f a2e[4];
        {
            const float* lr = &lds[wave][1][l16][0];
#pragma unroll
            for (int kk2 = 0; kk2 < 4; ++kk2) {
                a2e[kk2].x = lr[kk2 * 4 + half * 2];
                a2e[kk2].y = lr[kk2 * 4 + half * 2 + 1];
            }
        }
#pragma unroll
        for (int nt = 0; nt < 4; ++nt) {
#pragma unroll
            for (int kk2 = 0; kk2 < 4; ++kk2) {
                acc[1][nt] = __builtin_amdgcn_wmma_f32_16x16x4_f32(
                                 false, a2e[kk2], false, vfrag[nt * 4 + kk2],
                                 (short)0, acc[1][nt], false, false);
            }
        }
    }

    // ======= divisive normalization + store (both M-tiles) ==================
#pragma unroll
    for (int t = 0; t < 2; ++t) {
#pragma unroll
        for (int v = 0; v < 8; ++v) {
            float s = ssq[t][v];
            s += __shfl_xor(s, 1, 32);
            s += __shfl_xor(s, 2, 32);
            s += __shfl_xor(s, 4, 32);
            s += __shfl_xor(s, 8, 32);                 // stays within 16-lane half
            float pool  = sqrtf(s + 1e-20f);
            float scale = 1.0f / (SIGMA + pool);
#pragma unroll
            for (int nt = 0; nt < 4; ++nt) acc[t][nt][v] *= scale;
        }
#pragma unroll
        for (int nt = 0; nt < 4; ++nt)
#pragma unroll
            for (int v = 0; v < 8; ++v)
                Oh[(size_t)(qb0 + t * 16 + v + half * 8) * DD + nt * 16 + l16] =
                    acc[t][nt][v];
    }
}

// ---------------------------------------------------------------------------
extern "C" void kernel_launch(void* const* d_in, const int* in_sizes, int n_in,
                              void* d_out, int out_size, void* d_ws,
                              size_t ws_size, hipStream_t stream) {
    const float* Q    = (const float*)d_in[0];
    const float* K    = (const float*)d_in[1];
    const float* V    = (const float*)d_in[2];
    const float* gain = (const float*)d_in[3];
    const float* bias = (const float*)d_in[4];
    float* out   = (float*)d_out;
    float* rnorm = (float*)d_ws;                       // H*T floats = 128 KB

    knorm_kernel<<<(BH * TT + 255) / 256, 256, 0, stream>>>(K, rnorm);

    const int waves  = BH * (TT / 32);                 // 1024 query blocks
    const int blocks = waves / WPB;                    // 256 blocks
    lif_attn_kernel<<<blocks, WPB * 32, 0, stream>>>(Q, K, V, gain, bias,
                                                     rnorm, out);
}